// InverseHaarWaveletTransform3D_4715874091394
// MI455X (gfx1250) — compile-verified
//
#include <hip/hip_runtime.h>

typedef __attribute__((ext_vector_type(2))) float v2f;
typedef __attribute__((ext_vector_type(8))) float v8f;

// Input  coeffs: [b=4][s=8][c=8][t=17][h=128][w=128]  (channel dim 64 = s-major)
// Output y:      [b=4][c=8][ot=33][oh=256][ow=256]
#define T_STRIDE  (128 * 128)              // 16384
#define C_STRIDE  (17 * T_STRIDE)          // 278528
#define SB_STRIDE (8 * C_STRIDE)           // 2228224 (subband stride)
#define B_STRIDE  (8 * SB_STRIDE)

__global__ __launch_bounds__(256) void ihaar3d_wmma(const float* __restrict__ in,
                                                    float* __restrict__ out) {
  const int lane = threadIdx.x & 31;
  const int gw   = blockIdx.x * 8 + (threadIdx.x >> 5); // one wave per (b,c,t,h)
  const int h = gw & 127;
  int tmp = gw >> 7;
  const int t = tmp % 17;  tmp /= 17;
  const int c = tmp & 7;
  const int b = tmp >> 3;

  const int half = lane >> 4;   // lane half selects K-pair (B) / K-pair (A) / store tile
  const int lv   = lane & 15;

  // A matrix (16x4 f32): lane l holds A[M = l&15][K = half*2 + j].
  // Rows 8..15 duplicate rows 0..7 so D rows 8..15 mirror 0..7 (used for tile B stores).
  const int o = lane & 7;                  // output position (i<<2)|(j<<1)|k
  v2f a1, a2;
  {
    const int k0 = half * 2;
    a1.x = (__popc(o & (k0 + 0)) & 1) ? -0.3536f : 0.3536f;
    a1.y = (__popc(o & (k0 + 1)) & 1) ? -0.3536f : 0.3536f;
    a2.x = (__popc(o & (k0 + 4)) & 1) ? -0.3536f : 0.3536f;
    a2.y = (__popc(o & (k0 + 5)) & 1) ? -0.3536f : 0.3536f;
  }

  const float* inb = in + b * B_STRIDE + c * C_STRIDE + t * T_STRIDE + h * 128;
  const int sLo = half * 2;                // B layout: lane half picks K 0/1 vs 2/3
  const int obc = (b * 8 + c) * 33;        // output (b,c) frame base

  for (int wc = 0; wc < 4; ++wc) {
    const int w0 = wc * 32;
    const int wA = w0 + lv;                // tile A voxel for this lane's B column
    const int wB = wA + 16;                // tile B voxel

    if (wc < 3) __builtin_prefetch(inb + sLo * SB_STRIDE + wA + 32, 0, 0);

    v2f bA1, bA2, bB1, bB2;                // B (4x16 f32): coalesced b32 loads
    bA1.x = inb[(sLo + 0) * SB_STRIDE + wA];
    bA1.y = inb[(sLo + 1) * SB_STRIDE + wA];
    bA2.x = inb[(sLo + 4) * SB_STRIDE + wA];
    bA2.y = inb[(sLo + 5) * SB_STRIDE + wA];
    bB1.x = inb[(sLo + 0) * SB_STRIDE + wB];
    bB1.y = inb[(sLo + 1) * SB_STRIDE + wB];
    bB2.x = inb[(sLo + 4) * SB_STRIDE + wB];
    bB2.y = inb[(sLo + 5) * SB_STRIDE + wB];

    v8f cA = {};
    v8f cB = {};
    // K=8 via two chained 16x16x4 f32 WMMAs (subbands 0-3, then 4-7)
    cA = __builtin_amdgcn_wmma_f32_16x16x4_f32(false, a1, false, bA1, (short)0, cA, false, false);
    cA = __builtin_amdgcn_wmma_f32_16x16x4_f32(false, a2, false, bA2, (short)0, cA, false, false);
    cB = __builtin_amdgcn_wmma_f32_16x16x4_f32(false, a1, false, bB1, (short)0, cB, false, false);
    cB = __builtin_amdgcn_wmma_f32_16x16x4_f32(false, a2, false, bB2, (short)0, cB, false, false);

    // Lanes 0-15 store tile A (D rows 0-7); lanes 16-31 store tile B via the
    // duplicated rows 8-15 -> every lane stores voxel (w0 + lane).
    v8f d;
#pragma unroll
    for (int r = 0; r < 8; ++r) d[r] = half ? cB[r] : cA[r];

    const int ow = 2 * (w0 + lane);
#pragma unroll
    for (int i = 0; i < 2; ++i) {
      const int ot = 2 * t + i - 1;        // causal: drop first reconstructed frame
      if (ot < 0) continue;                // uniform (t==0 && i==0)
#pragma unroll
      for (int j = 0; j < 2; ++j) {
        const int oh = 2 * h + j;
        const int r  = i * 4 + j * 2;      // k=0,1 adjacent in D and in memory
        float2 pr;
        pr.x = d[r];
        pr.y = d[r + 1];
        *reinterpret_cast<float2*>(out + ((obc + ot) * 256 + oh) * 256 + ow) = pr;
      }
    }
  }
}

extern "C" void kernel_launch(void* const* d_in, const int* in_sizes, int n_in,
                              void* d_out, int out_size, void* d_ws, size_t ws_size,
                              hipStream_t stream) {
  const float* in = (const float*)d_in[0];
  float* out = (float*)d_out;
  // waves = 4*8*17*128 = 69632 ; 8 waves per 256-thread block -> 8704 blocks
  ihaar3d_wmma<<<dim3(8704), dim3(256), 0, stream>>>(in, out);
}